// SIRSODE_naive_int_70188355551727
// MI455X (gfx1250) — compile-verified
//
#include <hip/hip_runtime.h>

// SIR ODE forward-Euler scan, T = out_size/4 steps.
// Latency-bound sequential recurrence: one lane of one wave32 runs the whole
// chain; per-step critical path is exactly  v_mul_f32 -> v_fma_f32.
// Intervention mask is hoisted into a 3-segment loop split (constant alpha
// per segment), so the inner loop has zero compare/select work.

__device__ __forceinline__ void sir_seg(float4* __restrict__ out4, int j0, int j1,
                                        float alpha, float beta, float gamma,
                                        float& S, float& I, float& R, float iv) {
#pragma unroll 4
  for (int j = j0; j < j1; ++j) {
    // Only state-dependent multiply on the loop-carried path:
    float t = S * I;
    // S' = S + gamma*R - alpha*S*I
    float nS = fmaf(-alpha, t, fmaf(gamma, R, S));
    // I' = I - beta*I + alpha*S*I
    float nI = fmaf(alpha, t, fmaf(-beta, I, I));
    // R' = R - gamma*R + beta*I
    float nR = fmaf(beta, I, fmaf(-gamma, R, R));
    S = nS;
    I = nI;
    R = nR;
    // 16B row store -> global_store_b128, fire-and-forget (STOREcnt),
    // never on the critical path.
    out4[j] = make_float4(S, I, R, iv);
  }
}

__global__ void sir_scan_kernel(const float* __restrict__ y0,
                                const float* __restrict__ params,
                                float* __restrict__ out, int T) {
  if (threadIdx.x != 0 || blockIdx.x != 0) return;

  float S  = y0[0];
  float I  = y0[1];
  float R  = y0[2];
  float iv = y0[3];  // intervention offset; stays constant in every output row

  float a0    = params[0];
  float beta  = params[1];
  float gamma = params[2];

  float4* out4 = (float4*)out;
  out4[0] = make_float4(S, I, R, iv);  // row 0 = y0

  if (iv != 0.0f) {
    // Window: jf >= 4+i && jf <= 9+i  ->  alpha = 0 inside, params[0] outside.
    float lo = 4.0f + iv;
    float hi = 9.0f + iv;
    int a = (int)fminf(fmaxf(ceilf(lo), 1.0f), (float)T);
    int b = (int)fminf(fmaxf(floorf(hi) + 1.0f, (float)a), (float)T);
    sir_seg(out4, 1, a, a0,   beta, gamma, S, I, R, iv);  // pre-window
    sir_seg(out4, a, b, 0.0f, beta, gamma, S, I, R, iv);  // window (alpha=0)
    sir_seg(out4, b, T, a0,   beta, gamma, S, I, R, iv);  // post-window
  } else {
    // Non-interventional: mask == 1 for all steps.
    sir_seg(out4, 1, T, a0, beta, gamma, S, I, R, iv);
  }
}

extern "C" void kernel_launch(void* const* d_in, const int* in_sizes, int n_in,
                              void* d_out, int out_size, void* d_ws, size_t ws_size,
                              hipStream_t stream) {
  (void)in_sizes; (void)n_in; (void)d_ws; (void)ws_size;
  const float* y0     = (const float*)d_in[0];  // [S0, I0, R0, i]
  const float* params = (const float*)d_in[1];  // [alpha, beta, gamma]
  float*       out    = (float*)d_out;          // [T, 4] flattened

  const int T = out_size / 4;

  // One wave32; only lane 0 carries the (inherently serial) recurrence.
  sir_scan_kernel<<<1, 32, 0, stream>>>(y0, params, out, T);
}